// FeatureExpansion_62182536512307
// MI455X (gfx1250) — compile-verified
//
#include <hip/hip_runtime.h>

typedef __attribute__((ext_vector_type(2))) float v2f;
typedef __attribute__((ext_vector_type(8))) float v8f;

#define NB   256      // batch
#define NW   100      // windows per row
#define NS   10       // stride / window length
#define SP   12       // padded K (rows 10,11 zero) for K=4 WMMA steps
#define ND   64       // features
#define NP   2016     // D*(D-1)/2 strict lower triangle
#define OUTW 4288     // per-window output floats: 4*64 + 2*2016
#define EPSV 1e-7f

__device__ __forceinline__ float add_eps(float x) {
    return x + (x < 0.0f ? -EPSV : EPSV);
}

__global__ __launch_bounds__(128)
void feat_expansion_kernel(const float* __restrict__ in, float* __restrict__ out) {
    __shared__ float xs[NS * ND];        // raw window        (640 f)
    __shared__ float diff[SP * ND];      // centered, K-padded(768 f)
    __shared__ float stdv[ND];           // per-feature std   (64 f)
    __shared__ float cov[ND * 65];       // cov accumulators, padded stride

    const int tid  = threadIdx.x;
    const int lane = tid & 31;
    const int wave = tid >> 5;
    const int half = lane >> 4;          // 0: lanes 0-15, 1: lanes 16-31
    const int l16  = lane & 15;

    const int bw = blockIdx.x;
    const int bi = bw / NW;
    const int wi = bw - bi * NW;

    // window is 640 contiguous floats in [B, W*S, D] layout
    const float* src = in + ((size_t)bi * (NW * NS) + (size_t)wi * NS) * ND;
    float* dst = out + (size_t)bw * OUTW;

    // ---- stage window into LDS, vectorized ----
    const float4* src4 = (const float4*)src;
    float4* xs4 = (float4*)xs;
    for (int i = tid; i < (NS * ND) / 4; i += 128) xs4[i] = src4[i];
    __syncthreads();

    // ---- per-feature stats (one thread per feature) ----
    if (tid < ND) {
        const int d = tid;
        float sum = 0.0f, dec = 0.0f;
        #pragma unroll
        for (int s = 0; s < NS; ++s) {
            float v = xs[s * ND + d];
            sum += v;
            dec += v * ((float)(s + 1) * (1.0f / 55.0f));  // linspace(1,10)/55
        }
        float mean = sum * (1.0f / NS);
        float var = 0.0f;
        #pragma unroll
        for (int s = 0; s < NS; ++s) {
            float dv = xs[s * ND + d] - mean;
            diff[s * ND + d] = dv;
            var += dv * dv;
        }
        diff[10 * ND + d] = 0.0f;   // K padding for 3x K=4 WMMA steps
        diff[11 * ND + d] = 0.0f;
        var *= (1.0f / NS);
        float sd = sqrtf(var);
        stdv[d] = sd;
        float z   = mean / add_eps(sd);
        float ret = xs[(NS - 1) * ND + d] / add_eps(xs[d]) - 1.0f;
        dst[d]          = sd;
        dst[ND + d]     = z;
        dst[2 * ND + d] = dec;
        dst[3 * ND + d] = ret;
    }
    __syncthreads();

    // ---- cov = diff^T (64xS) * diff (Sx64) via V_WMMA_F32_16X16X4_F32 ----
    // each wave owns one 16-row band (kt = wave), sweeps 4 column tiles.
    const int kt = wave;
    #pragma unroll
    for (int mt = 0; mt < 4; ++mt) {
        v8f c = {};
        #pragma unroll
        for (int s0 = 0; s0 < SP; s0 += 4) {
            v2f a, bf;
            // A 16x4 layout: VGPR0 = {K=s0 | K=s0+2}, VGPR1 = {K=s0+1 | K=s0+3}
            a.x  = diff[(s0 + half * 2 + 0) * ND + kt * 16 + l16];
            a.y  = diff[(s0 + half * 2 + 1) * ND + kt * 16 + l16];
            // B 4x16: rows of diff striped across lane halves
            bf.x = diff[(s0 + half * 2 + 0) * ND + mt * 16 + l16];
            bf.y = diff[(s0 + half * 2 + 1) * ND + mt * 16 + l16];
            c = __builtin_amdgcn_wmma_f32_16x16x4_f32(false, a, false, bf,
                                                      (short)0, c, false, false);
        }
        // C layout: VGPR v -> M = v + 8*half, N = l16
        #pragma unroll
        for (int v = 0; v < 8; ++v) {
            int R = kt * 16 + v + half * 8;
            int C = mt * 16 + l16;
            cov[R * 65 + C] = c[v];
        }
    }
    __syncthreads();

    // ---- packed strict-lower-triangle cov/corr, fully coalesced stores ----
    for (int idx = tid; idx < NP; idx += 128) {
        // invert idx = r*(r-1)/2 + c, c < r
        float fr = (1.0f + sqrtf(8.0f * (float)idx + 1.0f)) * 0.5f;
        int r = (int)fr;
        while (r * (r - 1) / 2 > idx) --r;
        while ((r + 1) * r / 2 <= idx) ++r;
        int cc = idx - r * (r - 1) / 2;
        float cv = cov[r * 65 + cc];
        dst[256 + idx] = cv * (1.0f / (NS - 1));                      // cov_unbiased
        float denom = stdv[r] * stdv[cc];
        dst[256 + NP + idx] = (cv * (1.0f / NS)) / add_eps(denom);    // corr
    }
}

extern "C" void kernel_launch(void* const* d_in, const int* in_sizes, int n_in,
                              void* d_out, int out_size, void* d_ws, size_t ws_size,
                              hipStream_t stream) {
    const float* in = (const float*)d_in[0];
    float* out = (float*)d_out;
    dim3 grid(NB * NW);   // 25600 windows, one block each
    dim3 block(128);      // 4 wave32 -> 4 tile-rows of the 64x64 cov
    hipLaunchKernelGGL(feat_expansion_kernel, grid, block, 0, stream, in, out);
}